// DRMatcher_17824114278904
// MI455X (gfx1250) — compile-verified
//
#include <hip/hip_runtime.h>
#include <hip/hip_bf16.h>
#include <float.h>

// Problem constants (from reference setup_inputs)
#define BATCH 16
#define QN    4000
#define NCLS  80
#define GTN   128
#define KTOP  10
#define TILE  256
#define EPSF  1e-8f

// ---------------------------------------------------------------------------
// CDNA5 async global->LDS path (gfx1250). Guarded so the file always compiles.
// ---------------------------------------------------------------------------
#if defined(__has_builtin)
#if __has_builtin(__builtin_amdgcn_global_load_async_to_lds_b128) && \
    __has_builtin(__builtin_amdgcn_s_wait_asynccnt)
#define HAVE_ASYNC_LDS 1
#endif
#endif

typedef int v4i_t __attribute__((vector_size(4 * sizeof(int))));

__device__ __forceinline__ void async_copy16(const float* gsrc, float* ldst) {
#ifdef HAVE_ASYNC_LDS
  __builtin_amdgcn_global_load_async_to_lds_b128(
      (__attribute__((address_space(1))) v4i_t*)(const_cast<float*>(gsrc)),
      (__attribute__((address_space(3))) v4i_t*)ldst,
      0, 0);
#else
  *(float4*)ldst = *(const float4*)gsrc;
#endif
}

__device__ __forceinline__ void wait_async_le1() {
#ifdef HAVE_ASYNC_LDS
  __builtin_amdgcn_s_wait_asynccnt(1);
  __asm__ volatile("" ::: "memory");
#endif
}
__device__ __forceinline__ void wait_async_le0() {
#ifdef HAVE_ASYNC_LDS
  __builtin_amdgcn_s_wait_asynccnt(0);
  __asm__ volatile("" ::: "memory");
#endif
}

// ---------------------------------------------------------------------------
// Kernel 1: one block per (batch b, gt g).
//   - computes cost row C[g, 0..Q) and writes it to out[b][0][g][:]
//   - computes sum of top-10 IoU -> dynamic_k (1..10)
//   - finds the dynamic_k-th smallest (cost, q) pair (lexicographic, matching
//     JAX stable argsort tie-break) -> threshold stored in workspace
// ---------------------------------------------------------------------------
__global__ __launch_bounds__(TILE) void cost_topk_kernel(
    const float* __restrict__ logits,   // [B,Q,C]
    const float* __restrict__ boxes,    // [B,Q,4] cxcywh
    const int*   __restrict__ labels,   // [B,G]
    const float* __restrict__ tboxes,   // [B,G,4] cxcywh
    float* __restrict__ out,            // [B,2,G,Q]
    float* __restrict__ thr_cost,       // [B,G]
    int*   __restrict__ thr_idx)        // [B,G]
{
  const int g   = blockIdx.x;
  const int b   = blockIdx.y;
  const int tid = threadIdx.x;

  __shared__ float s_box[2][TILE * 4];       // double-buffered box tiles
  __shared__ float s_iou[TILE][KTOP];        // per-thread sorted top-10 iou (desc)
  __shared__ float s_cst[TILE][KTOP];        // per-thread sorted bottom-10 cost (asc)
  __shared__ int   s_qix[TILE][KTOP];

  // Target box / label (uniform per block)
  const float4 tb  = *(const float4*)(tboxes + ((size_t)b * GTN + g) * 4);
  const int    lab = labels[b * GTN + g];
  const float tx1 = tb.x - 0.5f * tb.z, ty1 = tb.y - 0.5f * tb.w;
  const float tx2 = tb.x + 0.5f * tb.z, ty2 = tb.y + 0.5f * tb.w;
  const float areaT = (tx2 - tx1) * (ty2 - ty1);

  const float* bsrc = boxes  + (size_t)b * QN * 4;
  const float* lsrc = logits + (size_t)b * QN * NCLS + lab;
  float*       crow = out    + ((size_t)b * 2 * GTN + g) * (size_t)QN;

  // Local sorted lists
  float liou[KTOP]; float lcst[KTOP]; int lqi[KTOP];
#pragma unroll
  for (int n = 0; n < KTOP; ++n) { liou[n] = -FLT_MAX; lcst[n] = FLT_MAX; lqi[n] = 0x7fffffff; }

  const int NT = (QN + TILE - 1) / TILE;

  // Prime pipeline: async-stage tile 0 into LDS buffer 0.
  if (tid < QN) async_copy16(bsrc + (size_t)tid * 4, &s_box[0][tid * 4]);

  for (int t = 0; t < NT; ++t) {
    const int  buf     = t & 1;
    const bool havnext = (t + 1) < NT;
    if (havnext) {                                // stage next tile (async DMA)
      const int qn = (t + 1) * TILE + tid;
      if (qn < QN) async_copy16(bsrc + (size_t)qn * 4, &s_box[buf ^ 1][tid * 4]);
    }
    const int q = t * TILE + tid;
    // Strided logit gather overlaps with the in-flight async box load.
    float logit = 0.0f;
    if (q < QN) logit = lsrc[(size_t)q * NCLS];

    if (havnext) wait_async_le1(); else wait_async_le0();

    if (q < QN) {
      const float pcx = s_box[buf][tid * 4 + 0];
      const float pcy = s_box[buf][tid * 4 + 1];
      const float pw  = s_box[buf][tid * 4 + 2];
      const float ph  = s_box[buf][tid * 4 + 3];
      const float px1 = pcx - 0.5f * pw, py1 = pcy - 0.5f * ph;
      const float px2 = pcx + 0.5f * pw, py2 = pcy + 0.5f * ph;
      const float areaP = (px2 - px1) * (py2 - py1);

      // IoU / GIoU
      const float iw = fmaxf(fminf(px2, tx2) - fmaxf(px1, tx1), 0.0f);
      const float ih = fmaxf(fminf(py2, ty2) - fmaxf(py1, ty1), 0.0f);
      const float inter = iw * ih;
      const float uni   = areaP + areaT - inter;
      const float iou   = inter / (uni + EPSF);
      const float cw = fmaxf(fmaxf(px2, tx2) - fminf(px1, tx1), 0.0f);
      const float ch = fmaxf(fmaxf(py2, ty2) - fminf(py1, ty1), 0.0f);
      const float areaC = cw * ch;
      const float giou  = iou - (areaC - uni) / (areaC + EPSF);

      // L1 on cxcywh
      const float l1 = fabsf(pcx - tb.x) + fabsf(pcy - tb.y) +
                       fabsf(pw - tb.z) + fabsf(ph - tb.w);

      // Focal classification cost
      const float p    = 1.0f / (1.0f + expf(-logit));
      const float negc = 0.75f * p * p * (-log1pf(-p + EPSF));
      const float posc = 0.25f * (1.0f - p) * (1.0f - p) * (-logf(p + EPSF));
      const float ccls = posc - negc;

      const float cost = 5.0f * l1 + 2.0f * ccls - 2.0f * giou;
      crow[q] = cost;

      // Local top-10 IoU (descending), insertion with early-out
      if (iou > liou[KTOP - 1]) {
        float x = iou;
#pragma unroll
        for (int n = 0; n < KTOP; ++n) {
          if (x > liou[n]) { const float tmp = liou[n]; liou[n] = x; x = tmp; }
        }
      }
      // Local bottom-10 (cost, q) lexicographic ascending
      if (cost < lcst[KTOP - 1] ||
          (cost == lcst[KTOP - 1] && q < lqi[KTOP - 1])) {
        float xc = cost; int xi = q;
#pragma unroll
        for (int n = 0; n < KTOP; ++n) {
          const bool lt = (xc < lcst[n]) || (xc == lcst[n] && xi < lqi[n]);
          if (lt) {
            const float tc = lcst[n]; const int ti = lqi[n];
            lcst[n] = xc; lqi[n] = xi; xc = tc; xi = ti;
          }
        }
      }
    }
  }

  // Dump local lists to LDS
#pragma unroll
  for (int n = 0; n < KTOP; ++n) {
    s_iou[tid][n] = liou[n];
    s_cst[tid][n] = lcst[n];
    s_qix[tid][n] = lqi[n];
  }

  // Tree merge of sorted 10-lists: 256 -> 1
  for (int s = TILE / 2; s >= 1; s >>= 1) {
    __syncthreads();
    if (tid < s) {
      // merge IoU lists (descending), keep best 10
      {
        int i = 0, j = 0;
        float o[KTOP];
#pragma unroll
        for (int n = 0; n < KTOP; ++n) {
          const float av = s_iou[tid][i < KTOP ? i : KTOP - 1];
          const float bv = s_iou[tid + s][j < KTOP ? j : KTOP - 1];
          const bool ta = (j >= KTOP) || (i < KTOP && av >= bv);
          o[n] = ta ? av : bv;
          if (ta) ++i; else ++j;
        }
#pragma unroll
        for (int n = 0; n < KTOP; ++n) s_iou[tid][n] = o[n];
      }
      // merge (cost, idx) lists (lexicographic ascending), keep best 10
      {
        int i = 0, j = 0;
        float oc[KTOP]; int oi[KTOP];
#pragma unroll
        for (int n = 0; n < KTOP; ++n) {
          const int ia = i < KTOP ? i : KTOP - 1;
          const int jb = j < KTOP ? j : KTOP - 1;
          const float ac = s_cst[tid][ia];     const int ai = s_qix[tid][ia];
          const float bc = s_cst[tid + s][jb]; const int bi = s_qix[tid + s][jb];
          const bool ta = (j >= KTOP) ||
                          (i < KTOP && (ac < bc || (ac == bc && ai < bi)));
          oc[n] = ta ? ac : bc; oi[n] = ta ? ai : bi;
          if (ta) ++i; else ++j;
        }
#pragma unroll
        for (int n = 0; n < KTOP; ++n) { s_cst[tid][n] = oc[n]; s_qix[tid][n] = oi[n]; }
      }
    }
  }
  __syncthreads();

  if (tid == 0) {
    float sum = 0.0f;
#pragma unroll
    for (int n = 0; n < KTOP; ++n) sum += s_iou[0][n];
    int k = (int)sum;                 // truncation, matches .astype(int32)
    if (k < 1) k = 1;
    if (k > KTOP) k = KTOP;
    thr_cost[b * GTN + g] = s_cst[0][k - 1];
    thr_idx [b * GTN + g] = s_qix[0][k - 1];
  }
}

// ---------------------------------------------------------------------------
// Kernel 2: one thread per anchor q.
//   matching[g,q] = (cost,q) <= threshold_pair[g]  (lexicographic)
//   then per-anchor conflict resolution via first-argmin over g.
// ---------------------------------------------------------------------------
__global__ __launch_bounds__(TILE) void assign_kernel(
    float* __restrict__ out,            // [B,2,G,Q] (reads cost plane, writes matching plane)
    const float* __restrict__ thr_cost, // [B,G]
    const int*   __restrict__ thr_idx)  // [B,G]
{
  const int b = blockIdx.y;
  const int q = blockIdx.x * TILE + threadIdx.x;

  __shared__ float s_tc[GTN];
  __shared__ int   s_ti[GTN];
  if (threadIdx.x < GTN) {
    s_tc[threadIdx.x] = thr_cost[b * GTN + threadIdx.x];
    s_ti[threadIdx.x] = thr_idx [b * GTN + threadIdx.x];
  }
  __syncthreads();
  if (q >= QN) return;

  const float* cplane = out + ((size_t)b * 2 * GTN) * (size_t)QN;
  float*       mplane = out + ((size_t)b * 2 * GTN + GTN) * (size_t)QN;

  unsigned mask[GTN / 32] = {0u, 0u, 0u, 0u};
  int   count = 0;
  float minc  = FLT_MAX;
  int   ming  = 0;

#pragma unroll 4
  for (int g = 0; g < GTN; ++g) {
    const float c = cplane[(size_t)g * QN + q];
    const bool sel = (c < s_tc[g]) || (c == s_tc[g] && q <= s_ti[g]);
    if (sel) { mask[g >> 5] |= (1u << (g & 31)); ++count; }
    if (c < minc) { minc = c; ming = g; }     // first occurrence of minimum
  }

  const bool conflict = (count > 1);
#pragma unroll 4
  for (int g = 0; g < GTN; ++g) {
    float v;
    if (conflict) v = (g == ming) ? 1.0f : 0.0f;
    else          v = ((mask[g >> 5] >> (g & 31)) & 1u) ? 1.0f : 0.0f;
    mplane[(size_t)g * QN + q] = v;
  }
}

// ---------------------------------------------------------------------------
extern "C" void kernel_launch(void* const* d_in, const int* in_sizes, int n_in,
                              void* d_out, int out_size, void* d_ws, size_t ws_size,
                              hipStream_t stream) {
  (void)in_sizes; (void)n_in; (void)out_size; (void)ws_size;
  const float* logits = (const float*)d_in[0];   // [16,4000,80] f32
  const float* boxes  = (const float*)d_in[1];   // [16,4000,4]  f32
  const int*   labels = (const int*)  d_in[2];   // [16,128]     i32
  const float* tboxes = (const float*)d_in[3];   // [16,128,4]   f32
  float* out = (float*)d_out;                    // [16,2,128,4000] f32

  float* thr_cost = (float*)d_ws;                                  // [16,128]
  int*   thr_idx  = (int*)((char*)d_ws + BATCH * GTN * sizeof(float)); // [16,128]

  dim3 grid1(GTN, BATCH);
  cost_topk_kernel<<<grid1, TILE, 0, stream>>>(logits, boxes, labels, tboxes,
                                               out, thr_cost, thr_idx);

  dim3 grid2((QN + TILE - 1) / TILE, BATCH);
  assign_kernel<<<grid2, TILE, 0, stream>>>(out, thr_cost, thr_idx);
}